// GATConv_DGL_33285996544498
// MI455X (gfx1250) — compile-verified
//
#include <hip/hip_runtime.h>
#include <hip/hip_bf16.h>

#define N_NODES 50000
#define N_EDGES 1600000
#define IN_DIM 256
#define OUT_DIM 128
#define NEG_SLOPE 0.2f

typedef __attribute__((ext_vector_type(2))) float v2f;
typedef __attribute__((ext_vector_type(8))) float v8f;

// ---------------------------------------------------------------------------
// Kernel 1: H = X @ W using V_WMMA_F32_16X16X4_F32 (fp32 WMMA, wave32).
// One wave computes a 16x16 tile of H; 8 waves/block cover 16 rows x 128 cols.
// grid.x = N_NODES/16 = 3125 (exact).
//
// Fragment layouts (CDNA5 ISA):
//   A 16x4 f32 : lane l -> M = l&15 ; VGPR v -> K = 2*(l>>4) + v
//   B 4x16 f32 : lane l -> N = l&15 ; VGPR v -> K = 2*(l>>4) + v
//   C 16x16 f32: lane l -> N = l&15 ; VGPR v -> M = v + 8*(l>>4)
// ---------------------------------------------------------------------------
__global__ __launch_bounds__(256) void gat_gemm_wmma(
    const float* __restrict__ X,   // [N_NODES, IN_DIM]
    const float* __restrict__ W,   // [IN_DIM, OUT_DIM]
    float* __restrict__ H)         // [N_NODES, OUT_DIM]
{
  const int lane = threadIdx.x & 31;
  const int wave = threadIdx.x >> 5;      // 0..7 -> N tile
  const int l16  = lane & 15;
  const int half = lane >> 4;

  const int mBase = blockIdx.x * 16;
  const int nBase = wave * 16;

  // A: row (mBase + l16), k = kk + 2*half + {0,1}  -> contiguous float2
  const float* aPtr = X + (size_t)(mBase + l16) * IN_DIM + 2 * half;
  // B: col (nBase + l16), k rows kk + 2*half + {0,1}
  const float* bPtr = W + (size_t)(2 * half) * OUT_DIM + nBase + l16;

  v8f acc = {};
#pragma unroll 4
  for (int kk = 0; kk < IN_DIM; kk += 4) {
    v2f a = *(const v2f*)(aPtr + kk);
    v2f b;
    b.x = bPtr[(size_t)kk * OUT_DIM];
    b.y = bPtr[(size_t)(kk + 1) * OUT_DIM];
    acc = __builtin_amdgcn_wmma_f32_16x16x4_f32(
        /*neg_a=*/false, a, /*neg_b=*/false, b,
        /*c_mod=*/(short)0, acc, /*reuse_a=*/false, /*reuse_b=*/false);
  }

  float* out = H + (size_t)(mBase + 8 * half) * OUT_DIM + nBase + l16;
#pragma unroll
  for (int v = 0; v < 8; ++v)
    out[(size_t)v * OUT_DIM] = acc[v];
}

// ---------------------------------------------------------------------------
// Kernel 2: el[i] = h[i] . attn_l ; er[i] = h[i] . attn_r  (warp per node)
// ---------------------------------------------------------------------------
__global__ __launch_bounds__(256) void node_logits(
    const float* __restrict__ H,
    const float* __restrict__ attn_l,
    const float* __restrict__ attn_r,
    float* __restrict__ el, float* __restrict__ er)
{
  const int lane = threadIdx.x & 31;
  const int node = blockIdx.x * (blockDim.x >> 5) + (threadIdx.x >> 5);
  if (node >= N_NODES) return;

  const float4 hv = *(const float4*)(H + (size_t)node * OUT_DIM + 4 * lane);
  const float4 al = *(const float4*)(attn_l + 4 * lane);
  const float4 ar = *(const float4*)(attn_r + 4 * lane);
  float sl = hv.x * al.x + hv.y * al.y + hv.z * al.z + hv.w * al.w;
  float sr = hv.x * ar.x + hv.y * ar.y + hv.z * ar.z + hv.w * ar.w;
#pragma unroll
  for (int off = 16; off > 0; off >>= 1) {
    sl += __shfl_down(sl, off, 32);
    sr += __shfl_down(sr, off, 32);
  }
  if (lane == 0) { el[node] = sl; er[node] = sr; }
}

// ---------------------------------------------------------------------------
// Kernel 0: init (harness poisons d_out/d_ws with 0xAA)
// ---------------------------------------------------------------------------
__global__ __launch_bounds__(256) void init_buffers(
    float* __restrict__ out, float* __restrict__ emax, float* __restrict__ esum)
{
  const int i = blockIdx.x * blockDim.x + threadIdx.x;
  if (i < N_NODES * OUT_DIM) out[i] = 0.0f;
  if (i < N_NODES) { emax[i] = -__builtin_inff(); esum[i] = 0.0f; }
}

// ---------------------------------------------------------------------------
// Kernel 3: e = leaky_relu(el[src] + er[dst]); segment max over dst
// ---------------------------------------------------------------------------
__global__ __launch_bounds__(256) void edge_score_max(
    const int* __restrict__ src, const int* __restrict__ dst,
    const float* __restrict__ el, const float* __restrict__ er,
    float* __restrict__ score, float* __restrict__ emax)
{
  const int e = blockIdx.x * blockDim.x + threadIdx.x;
  if (e >= N_EDGES) return;
  const float t = el[src[e]] + er[dst[e]];
  const float s = t > 0.0f ? t : NEG_SLOPE * t;
  score[e] = s;
  atomicMax(emax + dst[e], s);   // -> GLOBAL_ATOMIC_MAX_NUM_F32
}

// ---------------------------------------------------------------------------
// Kernel 4: e_exp = exp(e - emax[dst]); segment sum over dst (in-place buffer)
// ---------------------------------------------------------------------------
__global__ __launch_bounds__(256) void edge_exp_sum(
    const int* __restrict__ dst,
    float* __restrict__ score,           // in: score, out: e_exp
    const float* __restrict__ emax, float* __restrict__ esum)
{
  const int e = blockIdx.x * blockDim.x + threadIdx.x;
  if (e >= N_EDGES) return;
  const int d = dst[e];
  const float ex = __expf(score[e] - emax[d]);
  score[e] = ex;
  atomicAdd(esum + d, ex);
}

// ---------------------------------------------------------------------------
// Kernel 5: rst[dst] += (e_exp/e_sum[dst]) * h[src]   (warp per edge)
// Each lane owns 4 contiguous dims: float4 load (B128) + 4 f32 atomics.
// ---------------------------------------------------------------------------
__global__ __launch_bounds__(256) void aggregate(
    const int* __restrict__ src, const int* __restrict__ dst,
    const float* __restrict__ H,
    const float* __restrict__ eexp, const float* __restrict__ esum,
    float* __restrict__ out)
{
  const int lane = threadIdx.x & 31;
  const int e = blockIdx.x * (blockDim.x >> 5) + (threadIdx.x >> 5);
  if (e >= N_EDGES) return;

  const int s = src[e], d = dst[e];
  const float a = eexp[e] / esum[d];
  const float4 hv = *(const float4*)(H + (size_t)s * OUT_DIM + 4 * lane);
  float* o = out + (size_t)d * OUT_DIM + 4 * lane;
  atomicAdd(o + 0, a * hv.x);
  atomicAdd(o + 1, a * hv.y);
  atomicAdd(o + 2, a * hv.z);
  atomicAdd(o + 3, a * hv.w);
}

// ---------------------------------------------------------------------------
extern "C" void kernel_launch(void* const* d_in, const int* in_sizes, int n_in,
                              void* d_out, int out_size, void* d_ws, size_t ws_size,
                              hipStream_t stream) {
  (void)in_sizes; (void)n_in; (void)out_size; (void)ws_size;

  const float* features  = (const float*)d_in[0];  // [N_NODES, IN_DIM]
  const float* fc_weight = (const float*)d_in[1];  // [IN_DIM, OUT_DIM]
  const float* attn_l    = (const float*)d_in[2];  // [OUT_DIM]
  const float* attn_r    = (const float*)d_in[3];  // [OUT_DIM]
  const int*   src       = (const int*)d_in[4];    // [N_EDGES]
  const int*   dst       = (const int*)d_in[5];    // [N_EDGES]
  float*       out       = (float*)d_out;          // [N_NODES, OUT_DIM]

  // Workspace layout (floats): h | el | er | emax | esum | score  (~33 MB)
  float* h     = (float*)d_ws;
  float* el    = h + (size_t)N_NODES * OUT_DIM;
  float* er    = el + N_NODES;
  float* emax  = er + N_NODES;
  float* esum  = emax + N_NODES;
  float* score = esum + N_NODES;

  // 0: init output + segment buffers
  init_buffers<<<(N_NODES * OUT_DIM + 255) / 256, 256, 0, stream>>>(out, emax, esum);

  // 1: H = X @ W  (WMMA fp32)
  gat_gemm_wmma<<<N_NODES / 16, 256, 0, stream>>>(features, fc_weight, h);

  // 2: el / er per node
  node_logits<<<(N_NODES + 7) / 8, 256, 0, stream>>>(h, attn_l, attn_r, el, er);

  // 3: edge scores + segment max
  edge_score_max<<<(N_EDGES + 255) / 256, 256, 0, stream>>>(src, dst, el, er, score, emax);

  // 4: exp + segment sum
  edge_exp_sum<<<(N_EDGES + 255) / 256, 256, 0, stream>>>(dst, score, emax, esum);

  // 5: weighted scatter-sum aggregation (warp per edge)
  aggregate<<<(N_EDGES + 7) / 8, 256, 0, stream>>>(src, dst, h, score, esum, out);
}